// MCR2Variational_1803886265766
// MI455X (gfx1250) — compile-verified
//
#include <hip/hip_runtime.h>

#define N_TOT 131072
#define DDIM  128
#define NCLS  10
#define EPS_C 0.5f
#define MU_C  1.0f

typedef float v2f __attribute__((ext_vector_type(2)));
typedef float v8f __attribute__((ext_vector_type(8)));

// ---------------- workspace zero ----------------
__global__ void zero_ws_kernel(float* __restrict__ ws, int n) {
    int i = blockIdx.x * blockDim.x + threadIdx.x;
    if (i < n) ws[i] = 0.0f;
}

// ---------------- trPi = Pi.sum(axis=0) ----------------
__global__ __launch_bounds__(256) void trpi_kernel(const float* __restrict__ Pi,
                                                   float* __restrict__ trPi) {
    float cnt[NCLS] = {};
    const int stride = gridDim.x * blockDim.x;
    for (int n = blockIdx.x * blockDim.x + threadIdx.x; n < N_TOT; n += stride) {
        const float* row = Pi + (long long)n * NCLS;
#pragma unroll
        for (int c = 0; c < NCLS; ++c) cnt[c] += row[c];
    }
    __shared__ float red[256];
    for (int c = 0; c < NCLS; ++c) {
        red[threadIdx.x] = cnt[c];
        __syncthreads();
        for (int off = 128; off > 0; off >>= 1) {
            if (threadIdx.x < off) red[threadIdx.x] += red[threadIdx.x + off];
            __syncthreads();
        }
        if (threadIdx.x == 0) atomicAdd(&trPi[c], red[0]);
        __syncthreads();
    }
}

// ---------------- weighted grams M[j] = Z^T diag(w_j) Z (j<10: w=Pi[:,j]; j==10: w=1) ----------------
#define NB    32           // samples per LDS buffer (double-buffered)
#define PITCH 136          // 128 + 8: 2*PITCH % 64 == 16 -> half-wave rows on disjoint banks
#define CHUNK 1024         // samples per workgroup
#define NSC   (CHUNK / NB) // 32 sub-chunks

__global__ __launch_bounds__(256) void gram_kernel(const float* __restrict__ Z,
                                                   const float* __restrict__ Pi,
                                                   float* __restrict__ M) {
    __shared__ float zs[2][NB * PITCH];
    __shared__ float wls[2][NB];

    const int j    = blockIdx.y;                 // 0..10 (10 == unweighted gram)
    float* Mj      = M + (long long)j * (DDIM * DDIM);
    const int tid  = threadIdx.x;
    const int lane = tid & 31;
    const int wave = tid >> 5;
    const int m    = lane & 15;
    const int klo  = (lane >> 4) << 1;           // 0 for lanes 0-15, 2 for lanes 16-31
    const int d0   = wave * 16;

    const unsigned lds_base = (unsigned)(uintptr_t)&zs[0][0];

    // Issue NB x 128 fp32 tile as async DMA into LDS buffer `buf` (4x B128 per thread).
    // NOTE: the instruction's immediate offset adds to BOTH LDS and global addresses,
    // and our LDS pitch (136) != global row stride (128), so compute each pair explicitly.
    auto issue_tile = [&](int buf, long long r0) {
#pragma unroll
        for (int it = 0; it < 4; ++it) {
            int s  = tid + it * 256;             // float4 slot in [0, 1024)
            int rr = s >> 5;                     // row = (4s)/128
            int cc = (s & 31) << 2;              // col = (4s)%128
            unsigned long long gaddr =
                (unsigned long long)(uintptr_t)(Z + (r0 + rr) * DDIM + cc);
            unsigned laddr =
                lds_base + (unsigned)(((buf * NB + rr) * PITCH + cc) * 4);
            asm volatile("global_load_async_to_lds_b128 %0, %1, off"
                         :: "v"(laddr), "v"(gaddr)
                         : "memory");
        }
    };

    v8f acc[8] = {};

    const long long row0 = (long long)blockIdx.x * CHUNK;

    // prologue: fill buffer 0
    issue_tile(0, row0);
    if (tid < NB)
        wls[0][tid] = (j < NCLS) ? Pi[(row0 + tid) * NCLS + j] : 1.0f;

    for (int sc = 0; sc < NSC; ++sc) {
        const int cur = sc & 1;
        // own DMA complete -> LDS data visible; barrier -> all waves' DMA + prior compute done
        asm volatile("s_wait_asynccnt 0x0" ::: "memory");
        __syncthreads();

        if (sc + 1 < NSC) {
            const long long rn = row0 + (long long)(sc + 1) * NB;
            issue_tile(cur ^ 1, rn);             // overlaps with compute below
            if (tid < NB)
                wls[cur ^ 1][tid] = (j < NCLS) ? Pi[(rn + tid) * NCLS + j] : 1.0f;
        }

        const float* zb = &zs[cur][0];
        const float* wb = &wls[cur][0];
#pragma unroll
        for (int k = 0; k < NB; k += 4) {
            const float* rA = &zb[(k + klo) * PITCH];
            const float* rB = &zb[(k + klo + 1) * PITCH];
            const float w0 = wb[k + klo];
            const float w1 = wb[k + klo + 1];
            // A[m][k] = w_k * Z[n_k][d0+m]  (weight folded into A: 2 muls/k-step)
            v2f a;
            a.x = rA[d0 + m] * w0;
            a.y = rB[d0 + m] * w1;
#pragma unroll
            for (int t = 0; t < 8; ++t) {
                v2f b;                            // B[k][e] = Z[n_k][e]
                b.x = rA[t * 16 + m];
                b.y = rB[t * 16 + m];
                acc[t] = __builtin_amdgcn_wmma_f32_16x16x4_f32(
                    false, a, false, b, (short)0, acc[t], false, false);
            }
        }
    }

    // C/D layout: VGPR v -> M = v (lanes 0-15) or v+8 (lanes 16-31); N = lane%16
#pragma unroll
    for (int t = 0; t < 8; ++t)
#pragma unroll
        for (int v = 0; v < 8; ++v) {
            int mm = d0 + ((lane < 16) ? v : v + 8);
            int nn = t * 16 + m;
            atomicAdd(&Mj[mm * DDIM + nn], acc[t][v]);
        }
}

// ---------------- per-class: compress loss + UUt (WMMA) + reg residual ----------------
__global__ __launch_bounds__(256) void class_kernel(const float* __restrict__ Us,
                                                    const float* __restrict__ M,
                                                    const float* __restrict__ trPi,
                                                    float* __restrict__ accRc,
                                                    float* __restrict__ accReg) {
    const int j = blockIdx.x;
    const float* U  = Us + (long long)j * (DDIM * DDIM);
    const float* Mj = M  + (long long)j * (DDIM * DDIM);
    const int tid = threadIdx.x;
    __shared__ float red[256];

    // loss_compress: per-column squared norms -> log1p -> scaled sum
    float myv = 0.0f;
    if (tid < DDIM) {
        float cns = 0.0f;
        for (int d = 0; d < DDIM; ++d) {
            float u = U[d * DDIM + tid];
            cns += u * u;
        }
        myv = log1pf(((float)DDIM / (trPi[j] * EPS_C)) * cns);
    }
    red[tid] = myv;
    __syncthreads();
    for (int off = 128; off > 0; off >>= 1) {
        if (tid < off) red[tid] += red[tid + off];
        __syncthreads();
    }
    if (tid == 0) atomicAdd(accRc, (trPi[j] / (2.0f * (float)N_TOT)) * red[0]);
    __syncthreads();

    // UUt[d][e] = sum_k U[d][k]*U[e][k] via fp32 WMMA (Us is L2-resident, read direct)
    const int lane = tid & 31, wave = tid >> 5;
    const int m = lane & 15, klo = (lane >> 4) << 1, d0 = wave * 16;
    v8f acc[8] = {};
    for (int k = 0; k < DDIM; k += 4) {
        v2f a;
        a.x = U[(d0 + m) * DDIM + k + klo];
        a.y = U[(d0 + m) * DDIM + k + klo + 1];
#pragma unroll
        for (int t = 0; t < 8; ++t) {
            v2f b;                                // B[k][e] = U[e][k]
            b.x = U[(t * 16 + m) * DDIM + k + klo];
            b.y = U[(t * 16 + m) * DDIM + k + klo + 1];
            acc[t] = __builtin_amdgcn_wmma_f32_16x16x4_f32(
                false, a, false, b, (short)0, acc[t], false, false);
        }
    }
    float rsum = 0.0f;
#pragma unroll
    for (int t = 0; t < 8; ++t)
#pragma unroll
        for (int v = 0; v < 8; ++v) {
            int mm = d0 + ((lane < 16) ? v : v + 8);
            int nn = t * 16 + m;
            float diff = Mj[mm * DDIM + nn] - acc[t][v];
            rsum += diff * diff;
        }
    red[tid] = rsum;
    __syncthreads();
    for (int off = 128; off > 0; off >>= 1) {
        if (tid < off) red[tid] += red[tid + off];
        __syncthreads();
    }
    if (tid == 0) atomicAdd(accReg, red[0]);
}

// ---------------- Cholesky logdet of I + gram/512 + finalize ----------------
__global__ __launch_bounds__(128) void chol_kernel(const float* __restrict__ gram,
                                                   const float* __restrict__ accRc,
                                                   const float* __restrict__ accReg,
                                                   float* __restrict__ out) {
    __shared__ float s[DDIM * DDIM];             // 64 KB
    const int tid = threadIdx.x;
    for (int idx = tid; idx < DDIM * DDIM; idx += 128) {
        int i = idx >> 7, c = idx & 127;
        s[idx] = ((i == c) ? 1.0f : 0.0f) + gram[idx] * (1.0f / 512.0f);
    }
    __syncthreads();
    for (int k = 0; k < DDIM; ++k) {
        if (tid == k) s[k * DDIM + k] = sqrtf(s[k * DDIM + k]);
        __syncthreads();
        if (tid > k) s[tid * DDIM + k] /= s[k * DDIM + k];
        __syncthreads();
        if (tid > k) {
            float lik = s[tid * DDIM + k];
            for (int c = k + 1; c <= tid; ++c)
                s[tid * DDIM + c] -= lik * s[c * DDIM + k];
        }
        __syncthreads();
    }
    if (tid == 0) {
        float lr = 0.0f;
        for (int i = 0; i < DDIM; ++i) lr += logf(s[i * DDIM + i]);
        float rc = *accRc;
        float rg = 0.5f * MU_C * (*accReg);
        out[0] = -(lr - rc - rg);
        out[1] = lr;
        out[2] = rc;
        out[3] = rg;
    }
}

extern "C" void kernel_launch(void* const* d_in, const int* in_sizes, int n_in,
                              void* d_out, int out_size, void* d_ws, size_t ws_size,
                              hipStream_t stream) {
    (void)in_sizes; (void)n_in; (void)out_size; (void)ws_size;
    const float* Z  = (const float*)d_in[0];
    const float* Pi = (const float*)d_in[1];
    const float* Us = (const float*)d_in[2];
    float* out = (float*)d_out;

    float* ws     = (float*)d_ws;
    float* M      = ws;                          // [11][128][128]; M[10] = unweighted gram
    float* trPi   = ws + 11 * 16384;             // [10]
    float* accRc  = ws + 11 * 16384 + 12;
    float* accReg = ws + 11 * 16384 + 13;

    const int nzero = 11 * 16384 + 16;
    zero_ws_kernel<<<(nzero + 255) / 256, 256, 0, stream>>>(ws, nzero);
    trpi_kernel<<<256, 256, 0, stream>>>(Pi, trPi);
    dim3 g(N_TOT / CHUNK, 11);
    gram_kernel<<<g, 256, 0, stream>>>(Z, Pi, M);
    class_kernel<<<NCLS, 256, 0, stream>>>(Us, M, trPi, accRc, accReg);
    chol_kernel<<<1, 128, 0, stream>>>(M + 10 * 16384, accRc, accReg, out);
}